// MIMO_LSTM_31018253812118
// MI455X (gfx1250) — compile-verified
//
#include <hip/hip_runtime.h>
#include <math.h>
#include <stdint.h>

#define T_STEPS 128
#define IN_D    512
#define H_D     512
#define B_D     256
#define TAG_D   64
#define LM_DD   200
#define POS_DD  6
#define CAP_DD  3

typedef __attribute__((ext_vector_type(2))) float v2f;
typedef __attribute__((ext_vector_type(8))) float v8f;

#define WS_LD 36   // 64 x 32 W tile; stride 36 floats = 144B (16B-aligned rows, 16-bank spread)
#define XS_LD 64   // 32 x 64 X tile

__device__ __forceinline__ float sigm_(float x) { return 1.0f / (1.0f + __expf(-x)); }

// ---- async global->LDS staging (ASYNCcnt path; K%32==0, fully in-bounds) ---
__device__ __forceinline__ void async_wait_all_() {
  asm volatile("s_wait_asynccnt 0" ::: "memory");
}

__device__ __forceinline__ void stage_W_async(float* WsL, const float* W, int ldw,
                                              int row0, int ks, int tid) {
  int r = tid >> 2;                 // 0..63 (row)
  int q = (tid & 3) << 3;           // 0,8,16,24 (col, floats)
  unsigned long long src =
      (unsigned long long)(uintptr_t)(W + (size_t)(row0 + r) * ldw + ks + q);
  unsigned int dst = (unsigned int)(uintptr_t)(WsL + r * WS_LD + q);
  asm volatile("global_load_async_to_lds_b128 %0, %1, off\n\t"
               "global_load_async_to_lds_b128 %0, %1, off offset:16"
               :: "v"(dst), "v"(src) : "memory");
}

__device__ __forceinline__ void stage_X_async(float* XsL, const float* X, int ldx,
                                              int col0, int ks, int tid) {
  int kr = tid >> 3;                // 0..31 (k-row)
  int c  = (tid & 7) << 3;          // 0..56 (col, floats)
  unsigned long long src =
      (unsigned long long)(uintptr_t)(X + (size_t)(ks + kr) * ldx + col0 + c);
  unsigned int dst = (unsigned int)(uintptr_t)(XsL + kr * XS_LD + c);
  asm volatile("global_load_async_to_lds_b128 %0, %1, off\n\t"
               "global_load_async_to_lds_b128 %0, %1, off offset:16"
               :: "v"(dst), "v"(src) : "memory");
}

// ---- guarded VALU staging (K tails / transposed-A) --------------------------
__device__ __forceinline__ void stage_W(float* WsL, const float* W, int ldw,
                                        int row0, int ks, int K, int tid) {
  int r = tid >> 2;
  int q = (tid & 3) << 3;
  const float* src = W + (size_t)(row0 + r) * ldw + ks + q;
#pragma unroll
  for (int j = 0; j < 8; ++j) {
    int kk = ks + q + j;
    WsL[r * WS_LD + q + j] = (kk < K) ? src[j] : 0.0f;
  }
}

// transposed-A: A[m][k] = W[k*ldw + row0+m]
__device__ __forceinline__ void stage_Wt(float* WsL, const float* W, int ldw,
                                         int row0, int ks, int K, int tid) {
  int kk = tid >> 3;
  int mq = (tid & 7) << 3;
  const float* src = W + (size_t)(ks + kk) * ldw + row0 + mq;
  bool ok = (ks + kk) < K;
#pragma unroll
  for (int j = 0; j < 8; ++j)
    WsL[(mq + j) * WS_LD + kk] = ok ? src[j] : 0.0f;
}

__device__ __forceinline__ void stage_X(float* XsL, const float* X, int ldx,
                                        int col0, int ks, int K, int tid) {
  int kr = tid >> 3;
  int c  = (tid & 7) << 3;
  const float* src = X + (size_t)(ks + kr) * ldx + col0 + c;
  bool ok = (ks + kr) < K;
#pragma unroll
  for (int j = 0; j < 8; ++j)
    XsL[kr * XS_LD + c + j] = ok ? src[j] : 0.0f;
}

// ---- one 32-deep K-slab of WMMA for 2 tiles (same tile-row) ----------------
__device__ __forceinline__ void mma_slab(const float* WsL, const float* XsL,
                                         int lane, int tr, int tc0,
                                         v8f& acc0, v8f& acc1) {
  int lr = lane & 15;
  int half = lane >> 4;
  const float* wrow = WsL + (tr * 16 + lr) * WS_LD;
#pragma unroll
  for (int kk = 0; kk < 32; kk += 4) {
    v2f a;
    a.x = wrow[kk + 2 * half];
    a.y = wrow[kk + 2 * half + 1];
    const float* x0 = XsL + (kk + 2 * half) * XS_LD;
    const float* x1 = XsL + (kk + 2 * half + 1) * XS_LD;
    v2f b0, b1;
    b0.x = x0[tc0 * 16 + lr];       b0.y = x1[tc0 * 16 + lr];
    b1.x = x0[(tc0 + 1) * 16 + lr]; b1.y = x1[(tc0 + 1) * 16 + lr];
    acc0 = __builtin_amdgcn_wmma_f32_16x16x4_f32(false, a, false, b0, (short)0, acc0, false, false);
    acc1 = __builtin_amdgcn_wmma_f32_16x16x4_f32(false, a, false, b1, (short)0, acc1, false, false);
  }
}

// K loop, guarded VALU staging (supports transA and K tails)
__device__ __forceinline__ void gemm_pair(float* WsL, float* XsL,
    const float* W, int ldw, const float* X, int ldx, int K,
    int row0, int col0, int tid, int lane, int tr, int tc0,
    v8f& acc0, v8f& acc1, bool transA) {
  for (int ks = 0; ks < K; ks += 32) {
    if (transA) stage_Wt(WsL, W, ldw, row0, ks, K, tid);
    else        stage_W (WsL, W, ldw, row0, ks, K, tid);
    stage_X(XsL, X, ldx, col0, ks, K, tid);
    __syncthreads();
    mma_slab(WsL, XsL, lane, tr, tc0, acc0, acc1);
    __syncthreads();
  }
}

// K loop, async-DMA staging (K%32==0, row-major A)
__device__ __forceinline__ void gemm_pair_async(float* WsL, float* XsL,
    const float* W, int ldw, const float* X, int ldx, int K,
    int row0, int col0, int tid, int lane, int tr, int tc0,
    v8f& acc0, v8f& acc1) {
  for (int ks = 0; ks < K; ks += 32) {
    stage_W_async(WsL, W, ldw, row0, ks, tid);
    stage_X_async(XsL, X, ldx, col0, ks, tid);
    async_wait_all_();
    __syncthreads();
    mma_slab(WsL, XsL, lane, tr, tc0, acc0, acc1);
    __syncthreads();
  }
}

// ============================================================================
// Kernel: x[t] = inputs[t] + w_lmw^T @ lms[t] (+ pos/cap scalar tail)
__global__ void __launch_bounds__(256)
k_x(const float* __restrict__ inputs, const float* __restrict__ lms,
    const float* __restrict__ pos, const float* __restrict__ caps,
    const float* __restrict__ w_lmw, const float* __restrict__ w_posw,
    const float* __restrict__ w_capw, float* __restrict__ xout) {
  __shared__ float WsL[64 * WS_LD];
  __shared__ float XsL[32 * XS_LD];
  int t = blockIdx.z;
  int row0 = blockIdx.x * 64, col0 = blockIdx.y * 64;
  int tid = threadIdx.x, lane = tid & 31, wave = tid >> 5;
  int tr = wave >> 1, tc0 = (wave & 1) << 1;
  int lr = lane & 15, half = lane >> 4;
  const float* lms_t = lms + (size_t)t * LM_DD * B_D;
  const float* pos_t = pos + (size_t)t * POS_DD * B_D;
  const float* cap_t = caps + (size_t)t * CAP_DD * B_D;
  const float* in_t  = inputs + (size_t)t * IN_D * B_D;
  float* x_t = xout + (size_t)t * IN_D * B_D;

  v8f acc0 = {}, acc1 = {};
  gemm_pair(WsL, XsL, w_lmw, IN_D, lms_t, B_D, LM_DD, row0, col0,
            tid, lane, tr, tc0, acc0, acc1, /*transA=*/true);

#pragma unroll
  for (int j = 0; j < 2; ++j) {
    int n = col0 + (tc0 + j) * 16 + lr;
    v8f acc = j ? acc1 : acc0;
#pragma unroll
    for (int v = 0; v < 8; ++v) {
      int m = row0 + tr * 16 + half * 8 + v;
      float s = acc[v] + in_t[(size_t)m * B_D + n];
#pragma unroll
      for (int p = 0; p < POS_DD; ++p) s += w_posw[p * IN_D + m] * pos_t[p * B_D + n];
#pragma unroll
      for (int cc = 0; cc < CAP_DD; ++cc) s += w_capw[cc * IN_D + m] * cap_t[cc * B_D + n];
      x_t[(size_t)m * B_D + n] = s;
    }
  }
}

// Kernel: ii_all[t] = w_ii @ x[t]
__global__ void __launch_bounds__(256)
k_ii(const float* __restrict__ w_ii, const float* __restrict__ x,
     float* __restrict__ iiall) {
  __shared__ float WsL[64 * WS_LD];
  __shared__ float XsL[32 * XS_LD];
  int t = blockIdx.z;
  int row0 = blockIdx.x * 64, col0 = blockIdx.y * 64;
  int tid = threadIdx.x, lane = tid & 31, wave = tid >> 5;
  int tr = wave >> 1, tc0 = (wave & 1) << 1;
  int lr = lane & 15, half = lane >> 4;
  const float* x_t = x + (size_t)t * IN_D * B_D;
  float* d = iiall + (size_t)t * 4 * H_D * B_D;

  v8f acc0 = {}, acc1 = {};
  gemm_pair_async(WsL, XsL, w_ii, IN_D, x_t, B_D, IN_D, row0, col0,
                  tid, lane, tr, tc0, acc0, acc1);
#pragma unroll
  for (int j = 0; j < 2; ++j) {
    int n = col0 + (tc0 + j) * 16 + lr;
    v8f acc = j ? acc1 : acc0;
#pragma unroll
    for (int v = 0; v < 8; ++v) {
      int m = row0 + tr * 16 + half * 8 + v;
      d[(size_t)m * B_D + n] = acc[v];
    }
  }
}

// Kernel: t_add[t] = w_lmt @ lms[t] (+ pos/cap scalar tail)
__global__ void __launch_bounds__(256)
k_tadd(const float* __restrict__ w_lmt, const float* __restrict__ w_post,
       const float* __restrict__ w_capt, const float* __restrict__ lms,
       const float* __restrict__ pos, const float* __restrict__ caps,
       float* __restrict__ tadd) {
  __shared__ float WsL[64 * WS_LD];
  __shared__ float XsL[32 * XS_LD];
  int t = blockIdx.z;
  int row0 = blockIdx.x * 64, col0 = blockIdx.y * 64;
  int tid = threadIdx.x, lane = tid & 31, wave = tid >> 5;
  int tr = wave >> 1, tc0 = (wave & 1) << 1;
  int lr = lane & 15, half = lane >> 4;
  const float* lms_t = lms + (size_t)t * LM_DD * B_D;
  const float* pos_t = pos + (size_t)t * POS_DD * B_D;
  const float* cap_t = caps + (size_t)t * CAP_DD * B_D;
  float* d = tadd + (size_t)t * H_D * B_D;

  v8f acc0 = {}, acc1 = {};
  gemm_pair(WsL, XsL, w_lmt, LM_DD, lms_t, B_D, LM_DD, row0, col0,
            tid, lane, tr, tc0, acc0, acc1, false);
#pragma unroll
  for (int j = 0; j < 2; ++j) {
    int n = col0 + (tc0 + j) * 16 + lr;
    v8f acc = j ? acc1 : acc0;
#pragma unroll
    for (int v = 0; v < 8; ++v) {
      int m = row0 + tr * 16 + half * 8 + v;
      float s = acc[v];
#pragma unroll
      for (int p = 0; p < POS_DD; ++p) s += w_post[m * POS_DD + p] * pos_t[p * B_D + n];
#pragma unroll
      for (int cc = 0; cc < CAP_DD; ++cc) s += w_capt[m * CAP_DD + cc] * cap_t[cc * B_D + n];
      d[(size_t)m * B_D + n] = s;
    }
  }
}

// Kernel: step phase A — 7 fused GEMM slices (w_hi i/f/z/o, w_ti i/f/z) + gates
__global__ void __launch_bounds__(256)
k_gateA(const float* __restrict__ w_hi, const float* __restrict__ w_ti,
        const float* __restrict__ h, const float* __restrict__ tprev,
        const float* __restrict__ ii_t, const float* __restrict__ b_i,
        float* __restrict__ c, float* __restrict__ opart) {
  __shared__ float WsL[64 * WS_LD];
  __shared__ float XsL[32 * XS_LD];
  int row0 = blockIdx.x * 64, col0 = blockIdx.y * 64;
  int tid = threadIdx.x, lane = tid & 31, wave = tid >> 5;
  int tr = wave >> 1, tc0 = (wave & 1) << 1;
  int lr = lane & 15, half = lane >> 4;

  v8f acc[14] = {};
  for (int ks = 0; ks < H_D; ks += 32) {
    stage_X_async(XsL, h, B_D, col0, ks, tid);
#pragma unroll
    for (int s = 0; s < 4; ++s) {
      stage_W_async(WsL, w_hi + (size_t)s * H_D * H_D, H_D, row0, ks, tid);
      async_wait_all_();
      __syncthreads();
      mma_slab(WsL, XsL, lane, tr, tc0, acc[2 * s], acc[2 * s + 1]);
      __syncthreads();
    }
    stage_X_async(XsL, tprev, B_D, col0, ks, tid);
#pragma unroll
    for (int s = 0; s < 3; ++s) {
      stage_W_async(WsL, w_ti + (size_t)s * H_D * H_D, H_D, row0, ks, tid);
      async_wait_all_();
      __syncthreads();
      mma_slab(WsL, XsL, lane, tr, tc0, acc[8 + 2 * s], acc[9 + 2 * s]);
      __syncthreads();
    }
  }

  const size_t HB = (size_t)H_D * B_D;
#pragma unroll
  for (int j = 0; j < 2; ++j) {
    int n = col0 + (tc0 + j) * 16 + lr;
#pragma unroll
    for (int v = 0; v < 8; ++v) {
      int m = row0 + tr * 16 + half * 8 + v;
      size_t mn = (size_t)m * B_D + n;
      float pre_i = acc[0 + j][v] + acc[8 + j][v]  + ii_t[mn]          + b_i[m];
      float pre_f = acc[2 + j][v] + acc[10 + j][v] + ii_t[HB + mn]     + b_i[H_D + m];
      float pre_z = acc[4 + j][v] + acc[12 + j][v] + ii_t[2 * HB + mn] + b_i[2 * H_D + m];
      float iv = sigm_(pre_i);
      float fv = sigm_(pre_f);
      float zv = tanhf(pre_z);
      float cn = fv * c[mn] + iv * zv;
      c[mn] = cn;
      opart[mn] = acc[6 + j][v] + ii_t[3 * HB + mn] + b_i[3 * H_D + m];
    }
  }
}

// Kernel: step phase B — co = w_co@c_new, h = sigmoid(opart+co)*tanh(c_new)
__global__ void __launch_bounds__(256)
k_stepB(const float* __restrict__ w_co, const float* __restrict__ c,
        const float* __restrict__ opart, float* __restrict__ h) {
  __shared__ float WsL[64 * WS_LD];
  __shared__ float XsL[32 * XS_LD];
  int row0 = blockIdx.x * 64, col0 = blockIdx.y * 64;
  int tid = threadIdx.x, lane = tid & 31, wave = tid >> 5;
  int tr = wave >> 1, tc0 = (wave & 1) << 1;
  int lr = lane & 15, half = lane >> 4;

  v8f acc0 = {}, acc1 = {};
  gemm_pair_async(WsL, XsL, w_co, H_D, c, B_D, H_D, row0, col0,
                  tid, lane, tr, tc0, acc0, acc1);
#pragma unroll
  for (int j = 0; j < 2; ++j) {
    int n = col0 + (tc0 + j) * 16 + lr;
    v8f acc = j ? acc1 : acc0;
#pragma unroll
    for (int v = 0; v < 8; ++v) {
      int m = row0 + tr * 16 + half * 8 + v;
      size_t mn = (size_t)m * B_D + n;
      float ov = sigm_(opart[mn] + acc[v]);
      h[mn] = ov * tanhf(c[mn]);
    }
  }
}

// Kernel: step phase C — T_new = w_ht@h_new + b4 + t_add[t]  -> Ts[t]
__global__ void __launch_bounds__(256)
k_stepC(const float* __restrict__ w_ht, const float* __restrict__ h,
        const float* __restrict__ b_i, const float* __restrict__ tadd_t,
        float* __restrict__ Ts_t) {
  __shared__ float WsL[64 * WS_LD];
  __shared__ float XsL[32 * XS_LD];
  int row0 = blockIdx.x * 64, col0 = blockIdx.y * 64;
  int tid = threadIdx.x, lane = tid & 31, wave = tid >> 5;
  int tr = wave >> 1, tc0 = (wave & 1) << 1;
  int lr = lane & 15, half = lane >> 4;

  v8f acc0 = {}, acc1 = {};
  gemm_pair_async(WsL, XsL, w_ht, H_D, h, B_D, H_D, row0, col0,
                  tid, lane, tr, tc0, acc0, acc1);
#pragma unroll
  for (int j = 0; j < 2; ++j) {
    int n = col0 + (tc0 + j) * 16 + lr;
    v8f acc = j ? acc1 : acc0;
#pragma unroll
    for (int v = 0; v < 8; ++v) {
      int m = row0 + tr * 16 + half * 8 + v;
      size_t mn = (size_t)m * B_D + n;
      Ts_t[mn] = acc[v] + b_i[4 * H_D + m] + tadd_t[mn];
    }
  }
}

// Kernel: output heads + log_softmax over TAG dim, write (B,T,TAG) layouts
__global__ void __launch_bounds__(256)
k_y(const float* __restrict__ wyf, const float* __restrict__ byf,
    const float* __restrict__ wyc, const float* __restrict__ byc,
    const float* __restrict__ Ts, float* __restrict__ out) {
  __shared__ float WsL[64 * WS_LD];
  __shared__ float XsL[32 * XS_LD];
  __shared__ float redF[64 * 64];
  __shared__ float redC[64 * 64];
  __shared__ float lseF[64];
  __shared__ float lseC[64];
  int t = blockIdx.x;
  int col0 = blockIdx.y * 64;
  int tid = threadIdx.x, lane = tid & 31, wave = tid >> 5;
  int tr = wave >> 1, tc0 = (wave & 1) << 1;
  int lr = lane & 15, half = lane >> 4;
  const float* Ts_t = Ts + (size_t)t * H_D * B_D;

  v8f af0 = {}, af1 = {}, ac0 = {}, ac1 = {};
  gemm_pair_async(WsL, XsL, wyf, H_D, Ts_t, B_D, H_D, 0, col0, tid, lane, tr, tc0, af0, af1);
  gemm_pair_async(WsL, XsL, wyc, H_D, Ts_t, B_D, H_D, 0, col0, tid, lane, tr, tc0, ac0, ac1);

#pragma unroll
  for (int j = 0; j < 2; ++j) {
    int nl = (tc0 + j) * 16 + lr;
    v8f af = j ? af1 : af0;
    v8f ac = j ? ac1 : ac0;
#pragma unroll
    for (int v = 0; v < 8; ++v) {
      int m = tr * 16 + half * 8 + v;
      redF[m * 64 + nl] = af[v] + byf[m];
      redC[m * 64 + nl] = ac[v] + byc[m];
    }
  }
  __syncthreads();
  if (tid < 64) {
    float mx = -3.0e38f;
    for (int r = 0; r < TAG_D; ++r) mx = fmaxf(mx, redF[r * 64 + tid]);
    float s = 0.0f;
    for (int r = 0; r < TAG_D; ++r) s += __expf(redF[r * 64 + tid] - mx);
    lseF[tid] = mx + __logf(s);
  } else if (tid < 128) {
    int col = tid - 64;
    float mx = -3.0e38f;
    for (int r = 0; r < TAG_D; ++r) mx = fmaxf(mx, redC[r * 64 + col]);
    float s = 0.0f;
    for (int r = 0; r < TAG_D; ++r) s += __expf(redC[r * 64 + col] - mx);
    lseC[col] = mx + __logf(s);
  }
  __syncthreads();

  const size_t SZ = (size_t)B_D * T_STEPS * TAG_D;  // 2097152
#pragma unroll
  for (int j = 0; j < 2; ++j) {
    int nl = (tc0 + j) * 16 + lr;
    int n = col0 + nl;  // batch index
#pragma unroll
    for (int v = 0; v < 8; ++v) {
      int m = tr * 16 + half * 8 + v;  // tag index
      float yf = redF[m * 64 + nl];
      float yc = redC[m * 64 + nl];
      size_t ob = (size_t)n * (T_STEPS * TAG_D) + (size_t)t * TAG_D + m;
      out[ob]          = yf - lseF[nl];
      out[SZ + ob]     = yc - lseC[nl];
      out[2 * SZ + ob] = yf;
      out[3 * SZ + ob] = yc;
    }
  }
}

// Kernel: Ts (T,H,B) -> out (B,T,H), LDS-staged transpose
__global__ void __launch_bounds__(256)
k_ts(const float* __restrict__ Ts, float* __restrict__ out) {
  __shared__ float tile[64 * 65];
  int t = blockIdx.z;
  int h0 = blockIdx.x * 64, b0 = blockIdx.y * 64;
  int tid = threadIdx.x;
  const float* src = Ts + (size_t)t * H_D * B_D;
  const size_t OFF_TS = 4 * (size_t)B_D * T_STEPS * TAG_D;  // 8388608
#pragma unroll
  for (int i = 0; i < 16; ++i) {
    int idx = i * 256 + tid;
    int hr = idx >> 6, bc = idx & 63;
    tile[hr * 65 + bc] = src[(size_t)(h0 + hr) * B_D + b0 + bc];
  }
  __syncthreads();
#pragma unroll
  for (int i = 0; i < 16; ++i) {
    int idx = i * 256 + tid;
    int br = idx >> 6, hc = idx & 63;
    out[OFF_TS + (size_t)(b0 + br) * (T_STEPS * H_D) + (size_t)t * H_D + h0 + hc] =
        tile[hc * 65 + br];
  }
}

// ============================================================================
extern "C" void kernel_launch(void* const* d_in, const int* in_sizes, int n_in,
                              void* d_out, int out_size, void* d_ws, size_t ws_size,
                              hipStream_t stream) {
  const float* inputs   = (const float*)d_in[0];
  const float* lms      = (const float*)d_in[1];
  const float* pos      = (const float*)d_in[2];
  const float* caps     = (const float*)d_in[3];
  const float* h0       = (const float*)d_in[4];
  const float* c0       = (const float*)d_in[5];
  const float* t0       = (const float*)d_in[6];
  const float* w_ii     = (const float*)d_in[7];
  const float* w_hi     = (const float*)d_in[8];
  const float* w_ti     = (const float*)d_in[9];
  const float* w_co     = (const float*)d_in[10];
  const float* w_ht     = (const float*)d_in[11];
  const float* b_i      = (const float*)d_in[12];
  const float* w_y_fact = (const float*)d_in[13];
  const float* b_y_fact = (const float*)d_in[14];
  const float* w_y_cond = (const float*)d_in[15];
  const float* b_y_cond = (const float*)d_in[16];
  const float* w_lmw    = (const float*)d_in[17];
  const float* w_posw   = (const float*)d_in[18];
  const float* w_capw   = (const float*)d_in[19];
  const float* w_lmt    = (const float*)d_in[20];
  const float* w_post   = (const float*)d_in[21];
  const float* w_capt   = (const float*)d_in[22];
  float* out = (float*)d_out;

  const size_t HB = (size_t)H_D * B_D;                 // 131072
  float* ws    = (float*)d_ws;
  float* xbuf  = ws;                                   // T*IN*B   = 16,777,216
  float* iiall = xbuf + (size_t)T_STEPS * IN_D * B_D;  // T*4H*B   = 67,108,864
  float* tadd  = iiall + (size_t)T_STEPS * 4 * HB;     // T*H*B    = 16,777,216
  float* Tsbuf = tadd + (size_t)T_STEPS * HB;          // T*H*B    = 16,777,216
  float* hbuf  = Tsbuf + (size_t)T_STEPS * HB;
  float* cbuf  = hbuf + HB;
  float* opart = cbuf + HB;

  hipMemcpyAsync(hbuf, h0, HB * sizeof(float), hipMemcpyDeviceToDevice, stream);
  hipMemcpyAsync(cbuf, c0, HB * sizeof(float), hipMemcpyDeviceToDevice, stream);

  dim3 blk(256);
  // time-parallel precompute
  k_x   <<<dim3(IN_D / 64, B_D / 64, T_STEPS), blk, 0, stream>>>(
      inputs, lms, pos, caps, w_lmw, w_posw, w_capw, xbuf);
  k_ii  <<<dim3(4 * H_D / 64, B_D / 64, T_STEPS), blk, 0, stream>>>(w_ii, xbuf, iiall);
  k_tadd<<<dim3(H_D / 64, B_D / 64, T_STEPS), blk, 0, stream>>>(
      w_lmt, w_post, w_capt, lms, pos, caps, tadd);

  // sequential recurrence: 3 fused kernels per step
  dim3 sgrid(H_D / 64, B_D / 64);
  for (int t = 0; t < T_STEPS; ++t) {
    const float* tprev = (t == 0) ? t0 : (Tsbuf + (size_t)(t - 1) * HB);
    k_gateA<<<sgrid, blk, 0, stream>>>(w_hi, w_ti, hbuf, tprev,
                                       iiall + (size_t)t * 4 * HB, b_i, cbuf, opart);
    k_stepB<<<sgrid, blk, 0, stream>>>(w_co, cbuf, opart, hbuf);
    k_stepC<<<sgrid, blk, 0, stream>>>(w_ht, hbuf, b_i, tadd + (size_t)t * HB,
                                       Tsbuf + (size_t)t * HB);
  }

  // output heads + log_softmax, then Ts transpose
  k_y <<<dim3(T_STEPS, B_D / 64), blk, 0, stream>>>(
      w_y_fact, b_y_fact, w_y_cond, b_y_cond, Tsbuf, out);
  k_ts<<<dim3(H_D / 64, B_D / 64, T_STEPS), blk, 0, stream>>>(Tsbuf, out);

  (void)in_sizes; (void)n_in; (void)out_size; (void)ws_size;
}